// DetectionHead_88802743812247
// MI455X (gfx1250) — compile-verified
//
#include <hip/hip_runtime.h>
#include <hip/hip_bf16.h>
#include <stdint.h>

#define NLVL     8
#define N_ANCH   171008
#define TILE     1024
#define TILES    167        // 171008 / 1024
#define NBUCKET  2048
#define CAP      1024
#define MAXB     150
#define PADB     160        // 10 groups of 16 for WMMA tiles
#define NBATCH   32

// Level geometry (anchor bases are all multiples of TILE, so a tile never
// straddles two levels).
__constant__ int   g_base[NLVL + 1] = {0, 65536, 131072, 147456, 163840, 167936, 168960, 169984, 171008};
__constant__ int   g_hw[NLVL]       = {65536, 65536, 16384, 16384, 4096, 1024, 1024, 1024};
__constant__ int   g_wsh[NLVL]      = {8, 8, 7, 7, 6, 5, 5, 5};        // log2(width)
__constant__ float g_str[NLVL]      = {4.f, 4.f, 8.f, 8.f, 16.f, 32.f, 32.f, 32.f};
__constant__ float g_scale[NLVL]    = {27.5f, 35.5f, 55.5f, 71.5f, 111.5f, 191.5f, 255.5f, 319.5f}; // RF/2

struct Ptrs { const float* cls[8]; const float* reg[8]; };

typedef __attribute__((ext_vector_type(2))) float v2f;
typedef __attribute__((ext_vector_type(8))) float v8f;

// ---- CDNA5 async global->LDS staging ------------------------------------
__device__ __forceinline__ unsigned lds_off32(const void* p) {
  return (unsigned)(uintptr_t)p;       // flat LDS aperture keeps DS offset in addr[31:0]
}
__device__ __forceinline__ void async_load_b128(unsigned lds, unsigned long long gaddr) {
  asm volatile("global_load_async_to_lds_b128 %0, %1, off"
               :: "v"(lds), "v"(gaddr) : "memory");
}
__device__ __forceinline__ void wait_async0() {
  asm volatile("s_wait_asynccnt 0" ::: "memory");
}

// Map score in (0.5, 1] to a monotone bucket over its float bit pattern.
__device__ __forceinline__ unsigned score_bucket(float s) {
  unsigned b = __float_as_uint(s);
  if (b < 0x3F000000u) return 0u;
  unsigned k = (b - 0x3F000000u) >> 12;           // [0.5,1.0) -> 0..2047
  return k > 2047u ? 2047u : k;                   // s == 1.0 exactly
}

// ---- K1: score + per-image histogram ------------------------------------
__global__ __launch_bounds__(256)
void k_score_hist(Ptrs p, unsigned* __restrict__ hist) {
  const int tid = threadIdx.x;
  const int b   = blockIdx.y;
  const int a0  = blockIdx.x * TILE;
  int L = 0;
  #pragma unroll
  for (int i = 1; i < NLVL; ++i) if (a0 >= g_base[i]) L = i;
  const int hw = g_hw[L];
  const int o0 = a0 - g_base[L];
  const float* c0 = p.cls[L] + (size_t)b * 2u * (size_t)hw + o0;   // class-0 plane
  const float* c1 = c0 + hw;                                       // class-1 plane

  __shared__ float    s0[TILE];
  __shared__ float    s1[TILE];
  __shared__ unsigned shist[NBUCKET];
  for (int i = tid; i < NBUCKET; i += 256) shist[i] = 0u;

  async_load_b128(lds_off32(s0) + (unsigned)tid * 16u,
                  (unsigned long long)(uintptr_t)c0 + (unsigned long long)tid * 16u);
  async_load_b128(lds_off32(s1) + (unsigned)tid * 16u,
                  (unsigned long long)(uintptr_t)c1 + (unsigned long long)tid * 16u);
  wait_async0();
  __syncthreads();

  #pragma unroll
  for (int k = 0; k < TILE / 256; ++k) {
    const int i = tid + k * 256;
    const float d = s1[i] - s0[i];
    if (d > 0.f) {                                   // fg == (class 1 wins)
      const float s = 1.f / (1.f + expf(-d));        // == softmax max prob
      atomicAdd(&shist[score_bucket(s)], 1u);
    }
  }
  __syncthreads();
  unsigned* h = hist + (size_t)b * NBUCKET;
  for (int i = tid; i < NBUCKET; i += 256) {
    const unsigned v = shist[i];
    if (v) atomicAdd(&h[i], v);
  }
}

// ---- K2: find per-image threshold bucket for top-150 --------------------
__global__ void k_thresh(const unsigned* __restrict__ hist, unsigned* __restrict__ thr) {
  if (threadIdx.x != 0) return;
  const int b = blockIdx.x;
  const unsigned* h = hist + (size_t)b * NBUCKET;
  unsigned cum = 0, t = 0;
  for (int i = NBUCKET - 1; i >= 0; --i) {
    cum += h[i];
    if (cum >= MAXB) { t = (unsigned)i; break; }
  }
  thr[b] = t;
}

// ---- K3: collect candidates >= threshold --------------------------------
__global__ __launch_bounds__(256)
void k_collect(Ptrs p, const unsigned* __restrict__ thr,
               unsigned* __restrict__ cnt, unsigned long long* __restrict__ keys) {
  const int tid = threadIdx.x;
  const int b   = blockIdx.y;
  const int a0  = blockIdx.x * TILE;
  int L = 0;
  #pragma unroll
  for (int i = 1; i < NLVL; ++i) if (a0 >= g_base[i]) L = i;
  const int hw = g_hw[L];
  const int o0 = a0 - g_base[L];
  const float* c0 = p.cls[L] + (size_t)b * 2u * (size_t)hw + o0;
  const float* c1 = c0 + hw;

  __shared__ float s0[TILE];
  __shared__ float s1[TILE];
  async_load_b128(lds_off32(s0) + (unsigned)tid * 16u,
                  (unsigned long long)(uintptr_t)c0 + (unsigned long long)tid * 16u);
  async_load_b128(lds_off32(s1) + (unsigned)tid * 16u,
                  (unsigned long long)(uintptr_t)c1 + (unsigned long long)tid * 16u);
  wait_async0();
  __syncthreads();

  const unsigned tb = thr[b];
  unsigned long long* kb = keys + (size_t)b * CAP;
  #pragma unroll
  for (int k = 0; k < TILE / 256; ++k) {
    const int i = tid + k * 256;
    const float d = s1[i] - s0[i];
    if (d > 0.f) {
      const float s = 1.f / (1.f + expf(-d));
      if (score_bucket(s) >= tb) {
        const unsigned pos = atomicAdd(&cnt[b], 1u);
        if (pos < CAP) {
          const unsigned anchor = (unsigned)(a0 + i);
          // (score desc, index asc) sortable key; pad key 0 sorts last.
          kb[pos] = ((unsigned long long)__float_as_uint(s) << 32) |
                    (unsigned long long)(0xFFFFFFFFu - anchor);
        }
      }
    }
  }
}

// ---- K4: per-image sort (bitonic), box decode, WMMA area-sums, NMS ------
__global__ __launch_bounds__(256)
void k_sort_nms(Ptrs p, const unsigned* __restrict__ cnt,
                const unsigned long long* __restrict__ keys,
                float* __restrict__ out) {
  const int b = blockIdx.x;
  const int tid = threadIdx.x;
  __shared__ unsigned long long sk[CAP];
  __shared__ float bx1[PADB], by1[PADB], bx2[PADB], by2[PADB], bar[PADB], bsc[PADB];
  __shared__ float asum[PADB * 16];     // area[j] + area[i] panel for current i-group
  __shared__ int bval[PADB], bkeep[PADB];
  __shared__ int supp;

  const unsigned cb = cnt[b];
  const int C = (int)(cb < (unsigned)CAP ? cb : (unsigned)CAP);
  const unsigned long long* kb = keys + (size_t)b * CAP;
  for (int i = tid; i < CAP; i += 256) sk[i] = (i < C) ? kb[i] : 0ull;
  __syncthreads();

  // Bitonic sort, descending.
  for (unsigned k = 2; k <= (unsigned)CAP; k <<= 1) {
    for (unsigned j = k >> 1; j > 0; j >>= 1) {
      for (unsigned i = tid; i < (unsigned)CAP; i += 256) {
        const unsigned ixj = i ^ j;
        if (ixj > i) {
          const unsigned long long a = sk[i], c = sk[ixj];
          const bool desc = ((i & k) == 0);
          if (desc ? (a < c) : (a > c)) { sk[i] = c; sk[ixj] = a; }
        }
      }
      __syncthreads();
    }
  }

  if (tid < PADB) {
    const bool valid = (tid < C) && (tid < MAXB);
    float score = 0.f, x1 = 0.f, y1 = 0.f, x2 = 0.f, y2 = 0.f;
    if (valid) {
      const unsigned long long kk = sk[tid];
      score = __uint_as_float((unsigned)(kk >> 32));
      const unsigned anchor = 0xFFFFFFFFu - (unsigned)(kk & 0xFFFFFFFFull);
      int L = 0;
      #pragma unroll
      for (int i = 1; i < NLVL; ++i) if ((int)anchor >= g_base[i]) L = i;
      const int o   = (int)anchor - g_base[L];
      const int wsh = g_wsh[L];
      const float st = g_str[L];
      const float cx = (float)(o & ((1 << wsh) - 1)) * st + st * 0.5f;
      const float cy = (float)(o >> wsh) * st + st * 0.5f;
      const int hw = g_hw[L];
      const float* rp = p.reg[L] + (size_t)b * 4u * (size_t)hw + o;
      const float sc = g_scale[L];
      x1 = cx - rp[0]      * sc;
      y1 = cy - rp[hw]     * sc;
      x2 = cx + rp[2 * hw] * sc;
      y2 = cy + rp[3 * hw] * sc;
    }
    bx1[tid] = x1; by1[tid] = y1; bx2[tid] = x2; by2[tid] = y2;
    bar[tid] = (x2 - x1 + 1.f) * (y2 - y1 + 1.f);
    bsc[tid] = score;
    bval[tid] = valid ? 1 : 0;
    bkeep[tid] = 0;
  }
  __syncthreads();

  // Greedy NMS in score order, processed in 10 groups of 16 steps. For each
  // group, wave 0 builds the pairwise denominator base areasum[j][ii] =
  // area[j] + area[i0+ii] as a rank-2 outer product with
  // V_WMMA_F32_16X16X4_F32:  A(16x4) = [areaJ | 1 | 0 | 0],
  // B(4x16) = [1 ; areaI ; 0 ; 0]  => D = areaJ·1ᵀ + 1·areaIᵀ.
  for (int ig = 0; ig < PADB / 16; ++ig) {
    if (tid < 32) {                       // single fully-active wave: EXEC all ones
#if __has_builtin(__builtin_amdgcn_wmma_f32_16x16x4_f32)
      const int m  = tid & 15;
      const bool lo = tid < 16;           // lanes 0-15 hold K=0,1 ; lanes 16-31 K=2,3
      v2f bv;
      bv.x = lo ? 1.f : 0.f;              // B row K0 = ones
      bv.y = lo ? bar[ig * 16 + m] : 0.f; // B row K1 = areaI
      #pragma unroll
      for (int jb = 0; jb < PADB / 16; ++jb) {
        v2f av;
        av.x = lo ? bar[jb * 16 + m] : 0.f;   // A col K0 = areaJ
        av.y = lo ? 1.f : 0.f;                // A col K1 = ones
        v8f c = {};
        v8f d = __builtin_amdgcn_wmma_f32_16x16x4_f32(
            false, av, false, bv, (short)0, c, false, false);
        const int rbase = lo ? 0 : 8;         // D: lanes<16 rows 0-7, lanes>=16 rows 8-15
        #pragma unroll
        for (int r = 0; r < 8; ++r)
          asum[(jb * 16 + rbase + r) * 16 + m] = d[r];
      }
#else
      for (int j = tid; j < PADB; j += 32)
        for (int ii = 0; ii < 16; ++ii)
          asum[j * 16 + ii] = bar[j] + bar[ig * 16 + ii];
#endif
    }
    __syncthreads();
    for (int ii = 0; ii < 16; ++ii) {
      const int i = ig * 16 + ii;
      if (i >= MAXB) break;               // uniform across block
      if (tid == 0) supp = 0;
      __syncthreads();
      if (tid < i && bkeep[tid]) {
        const float xmin = fmaxf(bx1[tid], bx1[i]);
        const float ymin = fmaxf(by1[tid], by1[i]);
        const float xmax = fminf(bx2[tid], bx2[i]);
        const float ymax = fminf(by2[tid], by2[i]);
        const float inter = fmaxf(xmax - xmin, 0.f) * fmaxf(ymax - ymin, 0.f);
        const float iou = inter / (asum[tid * 16 + ii] - inter);  // WMMA-built denom
        if (iou > 0.3f) atomicOr(&supp, 1);   // NaN compares false, like jnp
      }
      __syncthreads();
      if (tid == 0) bkeep[i] = (bval[i] && !supp) ? 1 : 0;
      __syncthreads();
    }
  }

  // Outputs: scores[0,4800) classes[4800,9600) boxes[9600,28800) keep[28800,33600)
  if (tid < MAXB) {
    const int g = b * MAXB + tid;
    const bool kp = bkeep[tid] != 0;
    out[g]                     = kp ? bsc[tid] : 0.f;
    out[NBATCH * MAXB + g]     = kp ? 1.f : 0.f;     // fg class is always 1
    float* ob = out + 2 * NBATCH * MAXB + 4 * g;
    ob[0] = kp ? bx1[tid] : 0.f;
    ob[1] = kp ? by1[tid] : 0.f;
    ob[2] = kp ? bx2[tid] : 0.f;
    ob[3] = kp ? by2[tid] : 0.f;
    out[6 * NBATCH * MAXB + g] = kp ? 1.f : 0.f;     // keep mask
  }
}

extern "C" void kernel_launch(void* const* d_in, const int* in_sizes, int n_in,
                              void* d_out, int out_size, void* d_ws, size_t ws_size,
                              hipStream_t stream) {
  (void)in_sizes; (void)n_in; (void)out_size; (void)ws_size;
  Ptrs p;
  for (int i = 0; i < 8; ++i) {
    p.cls[i] = (const float*)d_in[i];
    p.reg[i] = (const float*)d_in[8 + i];
  }
  unsigned* hist = (unsigned*)d_ws;                       // 32*2048 u32
  unsigned* thr  = hist + NBATCH * NBUCKET;               // 32 u32
  unsigned* cntb = thr + NBATCH;                          // 32 u32
  unsigned long long* keyb =
      (unsigned long long*)((char*)d_ws + (size_t)(NBATCH * NBUCKET + 2 * NBATCH) * sizeof(unsigned));

  // Zero histograms/counters every launch (capture-safe, deterministic replay).
  hipMemsetAsync(d_ws, 0, (size_t)(NBATCH * NBUCKET + 2 * NBATCH) * sizeof(unsigned), stream);

  dim3 grid(TILES, NBATCH);
  k_score_hist<<<grid, 256, 0, stream>>>(p, hist);
  k_thresh   <<<NBATCH, 32, 0, stream>>>(hist, thr);
  k_collect  <<<grid, 256, 0, stream>>>(p, thr, cntb, keyb);
  k_sort_nms <<<NBATCH, 256, 0, stream>>>(p, cntb, keyb, (float*)d_out);
}